// HashEmbedder_1425929142827
// MI455X (gfx1250) — compile-verified
//
#include <hip/hip_runtime.h>
#include <stdint.h>

#define TABLE_SIZE   65536                 // 2^16 entries, embed dim 1
#define TABLE_BYTES  (TABLE_SIZE * 4)      // 256 KB, fits in 320 KB LDS/WGP
#define HASH_MASK    (TABLE_SIZE - 1)
#define PRIME1       2654435761u
#define BLOCK        1024                  // 32 waves -> fills one WGP
#define GRID         512

typedef float v4f __attribute__((ext_vector_type(4)));

// Async DMA: INST_OFFSET is added to BOTH the LDS dest address and the global
// source address (CDNA5 ISA), so one base pair + immediate
// offsets covers the whole table copy with zero per-load address arithmetic.
template <int OFF>
__device__ __forceinline__ void async_copy_b128(unsigned lds, unsigned long long gaddr) {
    asm volatile("global_load_async_to_lds_b128 %0, %1, off offset:%2"
                 :: "v"(lds), "v"(gaddr), "n"(OFF)
                 : "memory");
}

__global__ __launch_bounds__(BLOCK)
void HashEmbedder_kernel(const v4f* __restrict__ xv,
                         const float* __restrict__ table,
                         v4f* __restrict__ outv,
                         unsigned totalq)
{
    extern __shared__ float tab[];   // TABLE_BYTES dynamic LDS
    const unsigned tid = threadIdx.x;

    // ---- Phase 1: async DMA the 256 KB table into LDS (CDNA5 path) ----
    {
        // Low 32 bits of the flat shared-aperture address == LDS byte offset.
        unsigned lds = (unsigned)(size_t)(&tab[0]) + tid * 16u;
        unsigned long long g = (unsigned long long)(const void*)table + tid * 16u;
        // 256 KB / (1024 threads * 16 B) = 16 async b128 loads per thread.
        async_copy_b128<0 * BLOCK * 16>(lds, g);
        async_copy_b128<1 * BLOCK * 16>(lds, g);
        async_copy_b128<2 * BLOCK * 16>(lds, g);
        async_copy_b128<3 * BLOCK * 16>(lds, g);
        async_copy_b128<4 * BLOCK * 16>(lds, g);
        async_copy_b128<5 * BLOCK * 16>(lds, g);
        async_copy_b128<6 * BLOCK * 16>(lds, g);
        async_copy_b128<7 * BLOCK * 16>(lds, g);
        async_copy_b128<8 * BLOCK * 16>(lds, g);
        async_copy_b128<9 * BLOCK * 16>(lds, g);
        async_copy_b128<10 * BLOCK * 16>(lds, g);
        async_copy_b128<11 * BLOCK * 16>(lds, g);
        async_copy_b128<12 * BLOCK * 16>(lds, g);
        async_copy_b128<13 * BLOCK * 16>(lds, g);
        async_copy_b128<14 * BLOCK * 16>(lds, g);
        async_copy_b128<15 * BLOCK * 16>(lds, g);
        asm volatile("s_wait_asynccnt 0" ::: "memory");
    }
    __syncthreads();

    // ---- Phase 2: streaming hash + LDS gather, 4 points per thread/iter ----
    const unsigned stride = gridDim.x * BLOCK;
    for (unsigned q = blockIdx.x * BLOCK + tid; q < totalq; q += stride) {
        // points 4q..4q+3 : two NT b128 loads (read-once stream, keep L2 for table)
        v4f a = __builtin_nontemporal_load(xv + 2 * (size_t)q);
        v4f b = __builtin_nontemporal_load(xv + 2 * (size_t)q + 1);
        // grid_size = 2.0 exactly; x >= 0 so trunc cast == floor(x * 0.5)
        unsigned h0 = ((unsigned)(a.x * 0.5f) ^ ((unsigned)(a.y * 0.5f) * PRIME1)) & HASH_MASK;
        unsigned h1 = ((unsigned)(a.z * 0.5f) ^ ((unsigned)(a.w * 0.5f) * PRIME1)) & HASH_MASK;
        unsigned h2 = ((unsigned)(b.x * 0.5f) ^ ((unsigned)(b.y * 0.5f) * PRIME1)) & HASH_MASK;
        unsigned h3 = ((unsigned)(b.z * 0.5f) ^ ((unsigned)(b.w * 0.5f) * PRIME1)) & HASH_MASK;
        v4f r;
        r.x = tab[h0];                          // ds_load_b32 gathers
        r.y = tab[h1];
        r.z = tab[h2];
        r.w = tab[h3];
        __builtin_nontemporal_store(r, outv + q);   // NT b128 coalesced store
    }
}

extern "C" void kernel_launch(void* const* d_in, const int* in_sizes, int n_in,
                              void* d_out, int out_size, void* d_ws, size_t ws_size,
                              hipStream_t stream) {
    const v4f*   xv    = (const v4f*)d_in[0];   // [N,2] f32 -> N/2 float4
    const float* table = (const float*)d_in[1]; // [65536] f32
    v4f*         outv  = (v4f*)d_out;           // [N] f32 -> N/4 float4

    unsigned nPoints = (unsigned)((long long)in_sizes[0] / 2);
    unsigned totalq  = nPoints / 4;             // 4-point groups

    HashEmbedder_kernel<<<GRID, BLOCK, TABLE_BYTES, stream>>>(xv, table, outv, totalq);
}